// LegacyInteractiveNet_42099269435680
// MI455X (gfx1250) — compile-verified
//
#include <hip/hip_runtime.h>

typedef __attribute__((ext_vector_type(2))) float v2f;
typedef __attribute__((ext_vector_type(8))) float v8f;

#define DD 128
#define NN 256
#define BB 8
#define ROWS (BB * NN)        // 2048
#define WMSG_LD (2 * DD + 1)  // 257
#define WUPD_LD (2 * DD)      // 256

// ---------------------------------------------------------------------------
// Kernel 1: T = H * Wt^T,  S = H * Ws^T      (H: 2048x128, W slices: 128x128)
// One wave per 16x16 output tile; V_WMMA_F32_16X16X4_F32, K=128 -> 32 steps.
// ---------------------------------------------------------------------------
__global__ __launch_bounds__(256) void gemm_ts_kernel(
    const float* __restrict__ H, const float* __restrict__ Wmsg,
    float* __restrict__ T, float* __restrict__ S)
{
  const int wave = (blockIdx.x * 256 + threadIdx.x) >> 5;   // 0..1023
  const int lane = threadIdx.x & 31;
  const int mt = wave >> 3;        // 0..127
  const int nt = wave & 7;         // 0..7
  const int m0 = mt << 4, n0 = nt << 4;
  const int half = lane >> 4;
  const int ln = lane & 15;

  const float* __restrict__ arow = H + (size_t)(m0 + ln) * DD;
  const float* __restrict__ wrow = Wmsg + (size_t)(n0 + ln) * WMSG_LD;

  v8f accT = {};
  v8f accS = {};
  for (int k0 = 0; k0 < DD; k0 += 4) {
    const int ka = k0 + 2 * half;
    v2f a;  a.x  = arow[ka];          a.y  = arow[ka + 1];
    v2f bt; bt.x = wrow[ka];          bt.y = wrow[ka + 1];
    v2f bs; bs.x = wrow[DD + ka];     bs.y = wrow[DD + ka + 1];
    accT = __builtin_amdgcn_wmma_f32_16x16x4_f32(
        false, a, false, bt, (short)0, accT, false, false);
    accS = __builtin_amdgcn_wmma_f32_16x16x4_f32(
        false, a, false, bs, (short)0, accS, false, false);
  }
  #pragma unroll
  for (int r = 0; r < 8; ++r) {
    const int row = m0 + r + 8 * half;
    T[(size_t)row * DD + n0 + ln] = accT[r];
    S[(size_t)row * DD + n0 + ln] = accS[r];
  }
}

// ---------------------------------------------------------------------------
// Kernel 2a: G[b,i,j] = sum_e T[b,i,e] * S[b,j,e]   (batched T*S^T via WMMA)
// Supplies the only non-separable term of sum(pre^2) for the layernorm.
// ---------------------------------------------------------------------------
__global__ __launch_bounds__(256) void gemm_g_kernel(
    const float* __restrict__ T, const float* __restrict__ S,
    float* __restrict__ G)
{
  const int wave = (blockIdx.x * 256 + threadIdx.x) >> 5;   // 0..2047
  const int lane = threadIdx.x & 31;
  const int b  = wave >> 8;          // 0..7
  const int rm = wave & 255;
  const int it = rm >> 4;            // 0..15  (i tiles)
  const int jt = rm & 15;            // 0..15  (j tiles)
  const int m0 = it << 4, n0 = jt << 4;
  const int half = lane >> 4;
  const int ln = lane & 15;

  const float* __restrict__ arow = T + (size_t)(b * NN + m0 + ln) * DD;
  const float* __restrict__ brow = S + (size_t)(b * NN + n0 + ln) * DD;

  v8f acc = {};
  for (int k0 = 0; k0 < DD; k0 += 4) {
    const int ka = k0 + 2 * half;
    v2f a;  a.x = arow[ka]; a.y = arow[ka + 1];
    v2f bb; bb.x = brow[ka]; bb.y = brow[ka + 1];
    acc = __builtin_amdgcn_wmma_f32_16x16x4_f32(
        false, a, false, bb, (short)0, acc, false, false);
  }
  #pragma unroll
  for (int r = 0; r < 8; ++r) {
    const int i = m0 + r + 8 * half;
    G[(size_t)(b * NN + i) * NN + n0 + ln] = acc[r];
  }
}

// ---------------------------------------------------------------------------
// Kernel 2b: per-row statistics of S needed by the separable layernorm:
//   STATS[row] = { sum(s), sum(s^2), sum(s*wJ), sum(s*b_msg) }
// One wave per row; one-time butterfly reductions.
// ---------------------------------------------------------------------------
__global__ __launch_bounds__(256) void stats_s_kernel(
    const float* __restrict__ S, const float* __restrict__ Wmsg,
    const float* __restrict__ bmsg, float4* __restrict__ STATS)
{
  const int row  = (blockIdx.x * 256 + threadIdx.x) >> 5;   // 0..2047
  const int lane = threadIdx.x & 31;
  const int e0 = lane << 2;

  const float4 s4 = *(const float4*)(S + (size_t)row * DD + e0);
  float4 wj4;
  wj4.x = Wmsg[(size_t)(e0 + 0) * WMSG_LD + 2 * DD];
  wj4.y = Wmsg[(size_t)(e0 + 1) * WMSG_LD + 2 * DD];
  wj4.z = Wmsg[(size_t)(e0 + 2) * WMSG_LD + 2 * DD];
  wj4.w = Wmsg[(size_t)(e0 + 3) * WMSG_LD + 2 * DD];
  const float4 bm4 = *(const float4*)(bmsg + e0);

  float p1 = s4.x + s4.y + s4.z + s4.w;
  float p2 = s4.x * s4.x + s4.y * s4.y + s4.z * s4.z + s4.w * s4.w;
  float p3 = s4.x * wj4.x + s4.y * wj4.y + s4.z * wj4.z + s4.w * wj4.w;
  float p4 = s4.x * bm4.x + s4.y * bm4.y + s4.z * bm4.z + s4.w * bm4.w;
  #pragma unroll
  for (int off = 16; off >= 1; off >>= 1) {
    p1 += __shfl_xor(p1, off, 32);
    p2 += __shfl_xor(p2, off, 32);
    p3 += __shfl_xor(p3, off, 32);
    p4 += __shfl_xor(p4, off, 32);
  }
  if (lane == 0) {
    float4 st; st.x = p1; st.y = p2; st.z = p3; st.w = p4;
    STATS[row] = st;
  }
}

// ---------------------------------------------------------------------------
// Kernel 3: fused messages + aggregation with closed-form layernorm moments.
// One wave per (b,i); ZERO cross-lane ops in the j-loop:
//   mean  = (C1 + S1[j] + a*W1) / D
//   sumsq = C2 + SQ[j] + a^2*W2 + 2*(G[i,j] + SB[j]) + 2a*(Cw + SW[j])
// ---------------------------------------------------------------------------
__global__ __launch_bounds__(256) void msg_agg_kernel(
    const float* __restrict__ T, const float* __restrict__ S,
    const float* __restrict__ adj, const float* __restrict__ Wmsg,
    const float* __restrict__ bmsg, const float* __restrict__ gmsg,
    const float* __restrict__ betamsg, const float* __restrict__ G,
    const float4* __restrict__ STATS, float* __restrict__ AGG)
{
  const int wave = (blockIdx.x * 256 + threadIdx.x) >> 5;   // 0..2047
  const int lane = threadIdx.x & 31;
  const int b = wave >> 8;
  const int i = wave & 255;
  const int e0 = lane << 2;

  const float4 t4 = *(const float4*)(T + (size_t)(b * NN + i) * DD + e0);
  float4 wj4;
  wj4.x = Wmsg[(size_t)(e0 + 0) * WMSG_LD + 2 * DD];
  wj4.y = Wmsg[(size_t)(e0 + 1) * WMSG_LD + 2 * DD];
  wj4.z = Wmsg[(size_t)(e0 + 2) * WMSG_LD + 2 * DD];
  wj4.w = Wmsg[(size_t)(e0 + 3) * WMSG_LD + 2 * DD];
  const float4 bm4 = *(const float4*)(bmsg + e0);
  const float4 g4  = *(const float4*)(gmsg + e0);
  const float4 be4 = *(const float4*)(betamsg + e0);

  // c = t + b_msg (per-lane constant across the j loop)
  float4 c4;
  c4.x = t4.x + bm4.x; c4.y = t4.y + bm4.y;
  c4.z = t4.z + bm4.z; c4.w = t4.w + bm4.w;

  // One-time butterfly reductions for wave constants.
  float r1 = c4.x + c4.y + c4.z + c4.w;                                   // C1
  float r2 = c4.x * c4.x + c4.y * c4.y + c4.z * c4.z + c4.w * c4.w;       // C2
  float r3 = c4.x * wj4.x + c4.y * wj4.y + c4.z * wj4.z + c4.w * wj4.w;   // Cw
  float r4 = wj4.x + wj4.y + wj4.z + wj4.w;                               // W1
  float r5 = wj4.x * wj4.x + wj4.y * wj4.y + wj4.z * wj4.z + wj4.w * wj4.w; // W2
  #pragma unroll
  for (int off = 16; off >= 1; off >>= 1) {
    r1 += __shfl_xor(r1, off, 32);
    r2 += __shfl_xor(r2, off, 32);
    r3 += __shfl_xor(r3, off, 32);
    r4 += __shfl_xor(r4, off, 32);
    r5 += __shfl_xor(r5, off, 32);
  }
  const float C1 = r1, C2 = r2, Cw = r3, W1 = r4, W2 = r5;
  const float invD = 1.0f / DD;

  const float*  __restrict__ srow = S + (size_t)b * NN * DD;
  const float*  __restrict__ arow = adj + (size_t)(b * NN + i) * NN;
  const float*  __restrict__ grow = G + (size_t)(b * NN + i) * NN;
  const float4* __restrict__ strow = STATS + (size_t)b * NN;

  float4 acc = {0.f, 0.f, 0.f, 0.f};
  for (int j = 0; j < NN; ++j) {
    const float4 s4 = *(const float4*)(srow + (size_t)j * DD + e0);
    const float  aj = arow[j];
    const float  gij = grow[j];
    const float4 st = strow[j];     // {S1, SQ, SW, SB}

    const float mean  = (C1 + st.x + aj * W1) * invD;
    const float sumsq = C2 + st.y + aj * aj * W2
                      + 2.0f * (gij + st.w)
                      + 2.0f * aj * (Cw + st.z);
    const float var  = sumsq * invD - mean * mean;
    const float rinv = rsqrtf(var + 1e-5f);

    float4 p;
    p.x = c4.x + s4.x + aj * wj4.x;
    p.y = c4.y + s4.y + aj * wj4.y;
    p.z = c4.z + s4.z + aj * wj4.z;
    p.w = c4.w + s4.w + aj * wj4.w;

    float x0 = (p.x - mean) * rinv * g4.x + be4.x;
    float x1 = (p.y - mean) * rinv * g4.y + be4.y;
    float x2 = (p.z - mean) * rinv * g4.z + be4.z;
    float x3 = (p.w - mean) * rinv * g4.w + be4.w;
    const float m0 = x0 / (1.0f + __expf(-x0));
    const float m1 = x1 / (1.0f + __expf(-x1));
    const float m2 = x2 / (1.0f + __expf(-x2));
    const float m3 = x3 / (1.0f + __expf(-x3));

    const float keep = (j != i) ? 1.0f : 0.0f;
    acc.x += keep * m0;
    acc.y += keep * m1;
    acc.z += keep * m2;
    acc.w += keep * m3;
  }
  *(float4*)(AGG + (size_t)(b * NN + i) * DD + e0) = acc;
}

// ---------------------------------------------------------------------------
// Kernel 4: U = H * Wu1^T + AGG * Wu2^T + b_upd
// ---------------------------------------------------------------------------
__global__ __launch_bounds__(256) void gemm_upd_kernel(
    const float* __restrict__ H, const float* __restrict__ AGG,
    const float* __restrict__ Wupd, const float* __restrict__ bupd,
    float* __restrict__ U)
{
  const int wave = (blockIdx.x * 256 + threadIdx.x) >> 5;
  const int lane = threadIdx.x & 31;
  const int mt = wave >> 3;
  const int nt = wave & 7;
  const int m0 = mt << 4, n0 = nt << 4;
  const int half = lane >> 4;
  const int ln = lane & 15;

  const float* __restrict__ arow1 = H + (size_t)(m0 + ln) * DD;
  const float* __restrict__ arow2 = AGG + (size_t)(m0 + ln) * DD;
  const float* __restrict__ wrow = Wupd + (size_t)(n0 + ln) * WUPD_LD;

  v8f acc = {};
  for (int k0 = 0; k0 < DD; k0 += 4) {
    const int ka = k0 + 2 * half;
    v2f a1; a1.x = arow1[ka]; a1.y = arow1[ka + 1];
    v2f b1; b1.x = wrow[ka];  b1.y = wrow[ka + 1];
    acc = __builtin_amdgcn_wmma_f32_16x16x4_f32(
        false, a1, false, b1, (short)0, acc, false, false);
    v2f a2; a2.x = arow2[ka];     a2.y = arow2[ka + 1];
    v2f b2; b2.x = wrow[DD + ka]; b2.y = wrow[DD + ka + 1];
    acc = __builtin_amdgcn_wmma_f32_16x16x4_f32(
        false, a2, false, b2, (short)0, acc, false, false);
  }

  const float bias = bupd[n0 + ln];
  #pragma unroll
  for (int r = 0; r < 8; ++r) {
    const int row = m0 + r + 8 * half;
    U[(size_t)row * DD + n0 + ln] = acc[r] + bias;
  }
}

// ---------------------------------------------------------------------------
// Kernel 5: out = h + silu(layernorm(U, g_upd, beta_upd)). One wave per row.
// ---------------------------------------------------------------------------
__global__ __launch_bounds__(256) void finalize_kernel(
    const float* __restrict__ U, const float* __restrict__ H,
    const float* __restrict__ gupd, const float* __restrict__ betaupd,
    float* __restrict__ out)
{
  const int row = (blockIdx.x * 256 + threadIdx.x) >> 5;   // 0..2047
  const int lane = threadIdx.x & 31;
  const int e0 = lane << 2;

  const float4 u4 = *(const float4*)(U + (size_t)row * DD + e0);
  float ps  = u4.x + u4.y + u4.z + u4.w;
  float pss = u4.x * u4.x + u4.y * u4.y + u4.z * u4.z + u4.w * u4.w;
  #pragma unroll
  for (int off = 16; off >= 1; off >>= 1) {
    ps  += __shfl_xor(ps,  off, 32);
    pss += __shfl_xor(pss, off, 32);
  }
  const float mean = ps * (1.0f / DD);
  const float var  = pss * (1.0f / DD) - mean * mean;
  const float rinv = rsqrtf(var + 1e-5f);

  const float4 g4  = *(const float4*)(gupd + e0);
  const float4 be4 = *(const float4*)(betaupd + e0);
  const float4 h4  = *(const float4*)(H + (size_t)row * DD + e0);

  float x0 = (u4.x - mean) * rinv * g4.x + be4.x;
  float x1 = (u4.y - mean) * rinv * g4.y + be4.y;
  float x2 = (u4.z - mean) * rinv * g4.z + be4.z;
  float x3 = (u4.w - mean) * rinv * g4.w + be4.w;

  float4 o;
  o.x = h4.x + x0 / (1.0f + __expf(-x0));
  o.y = h4.y + x1 / (1.0f + __expf(-x1));
  o.z = h4.z + x2 / (1.0f + __expf(-x2));
  o.w = h4.w + x3 / (1.0f + __expf(-x3));
  *(float4*)(out + (size_t)row * DD + e0) = o;
}

extern "C" void kernel_launch(void* const* d_in, const int* in_sizes, int n_in,
                              void* d_out, int out_size, void* d_ws, size_t ws_size,
                              hipStream_t stream) {
  const float* h       = (const float*)d_in[0];
  const float* adj     = (const float*)d_in[1];
  const float* Wmsg    = (const float*)d_in[2];
  const float* bmsg    = (const float*)d_in[3];
  const float* gmsg    = (const float*)d_in[4];
  const float* betamsg = (const float*)d_in[5];
  const float* Wupd    = (const float*)d_in[6];
  const float* bupd    = (const float*)d_in[7];
  const float* gupd    = (const float*)d_in[8];
  const float* betaupd = (const float*)d_in[9];
  float* out = (float*)d_out;

  const size_t mat_bytes = (size_t)ROWS * DD * sizeof(float);        // 1 MB
  const size_t g_bytes   = (size_t)BB * NN * NN * sizeof(float);     // 2 MB
  char* ws = (char*)d_ws;
  float*  T     = (float*)(ws);
  float*  S     = (float*)(ws + mat_bytes);
  float*  AGG   = (float*)(ws + 2 * mat_bytes);
  float*  U     = (float*)(ws + 3 * mat_bytes);
  float*  G     = (float*)(ws + 4 * mat_bytes);
  float4* STATS = (float4*)(ws + 4 * mat_bytes + g_bytes);           // 32 KB

  gemm_ts_kernel<<<128, 256, 0, stream>>>(h, Wmsg, T, S);
  gemm_g_kernel<<<256, 256, 0, stream>>>(T, S, G);
  stats_s_kernel<<<256, 256, 0, stream>>>(S, Wmsg, bmsg, STATS);
  msg_agg_kernel<<<256, 256, 0, stream>>>(T, S, adj, Wmsg, bmsg, gmsg,
                                          betamsg, G, STATS, AGG);
  gemm_upd_kernel<<<128, 256, 0, stream>>>(h, AGG, Wupd, bupd, U);
  finalize_kernel<<<256, 256, 0, stream>>>(U, h, gupd, betaupd, out);
}